// FinalAttentionQKV_11158325035214
// MI455X (gfx1250) — compile-verified
//
#include <hip/hip_runtime.h>
#include <math.h>

#define BB 128
#define TT 2048
#define DD 128
#define HH 128

typedef __attribute__((ext_vector_type(2))) float v2f;
typedef __attribute__((ext_vector_type(8))) float v8f;

// ---------------------------------------------------------------------------
// K1: partial column sums of x over T (4 chunks of 512 rows per batch)
// ---------------------------------------------------------------------------
__global__ __launch_bounds__(256) void colsum_partial(const float* __restrict__ x,
                                                      float* __restrict__ partial) {
  const int b = blockIdx.x, chunk = blockIdx.y;
  const int g = threadIdx.x >> 7, d = threadIdx.x & 127;
  const int tbase = chunk * 512 + g * 256;
  const float* xb = x + ((b * TT + tbase) * DD) + d;
  float s = 0.f;
  for (int i = 0; i < 256; ++i) s += xb[i * DD];
  __shared__ float r[256];
  r[threadIdx.x] = s;
  __syncthreads();
  if (g == 0) partial[(b * 4 + chunk) * DD + d] = r[d] + r[128 + d];
}

__global__ __launch_bounds__(128) void colsum_final(const float* __restrict__ partial,
                                                    float* __restrict__ xmean) {
  const int b = blockIdx.x, d = threadIdx.x;
  float s = partial[(b * 4 + 0) * DD + d] + partial[(b * 4 + 1) * DD + d] +
            partial[(b * 4 + 2) * DD + d] + partial[(b * 4 + 3) * DD + d];
  xmean[b * DD + d] = s * (1.0f / (float)TT);
}

// ---------------------------------------------------------------------------
// 128x128x128 fp32 GEMM via V_WMMA_F32_16X16X4_F32.
// C[m,n] = sum_k A[m,k] * B'[k,n] + bias[n],
//   B'[k,n] = TRANS ? W[n*128+k] : W[k*128+n]
// One wave (32 threads) per 16x16 output tile; 32 K-steps of k=4.
// ---------------------------------------------------------------------------
template <bool TRANS>
__global__ __launch_bounds__(32) void gemm128_wmma(const float* __restrict__ A,
                                                   const float* __restrict__ W,
                                                   const float* __restrict__ bias,
                                                   float* __restrict__ C) {
  const int n0 = blockIdx.x * 16;
  const int m0 = blockIdx.y * 16;
  const int lane = threadIdx.x & 31;
  const int mrow = lane & 15;   // M for A frag, N for B/C frags
  const int hi = lane >> 4;     // half-wave selector

  v8f acc = {};
  for (int k0 = 0; k0 < 128; k0 += 4) {
    // A 16x4 frag: lane holds A[m0+mrow][k0+2*hi .. +1] (consecutive -> float2)
    v2f a = *(const v2f*)(A + (m0 + mrow) * 128 + k0 + 2 * hi);
    // B 4x16 frag: VGPR r holds row K = k0 + r + 2*hi, column N = n0 + mrow
    v2f bfrag;
    const int kb = k0 + 2 * hi;
    const int n = n0 + mrow;
    if (TRANS) {
      bfrag.x = W[n * 128 + kb + 0];
      bfrag.y = W[n * 128 + kb + 1];
    } else {
      bfrag.x = W[(kb + 0) * 128 + n];
      bfrag.y = W[(kb + 1) * 128 + n];
    }
    acc = __builtin_amdgcn_wmma_f32_16x16x4_f32(false, a, false, bfrag,
                                                (short)0, acc, false, false);
  }
  const int n = n0 + mrow;
  const float bb = bias ? bias[n] : 0.0f;
#pragma unroll
  for (int r = 0; r < 8; ++r) {
    const int m = m0 + r + 8 * hi;
    C[m * 128 + n] = acc[r] + bb;
  }
}

// ---------------------------------------------------------------------------
// dot0[b] = sum_h bk[h] * q[b,h]
// ---------------------------------------------------------------------------
__global__ __launch_bounds__(128) void dot0_kernel(const float* __restrict__ q,
                                                   const float* __restrict__ bk,
                                                   float* __restrict__ dot0) {
  const int b = blockIdx.x;
  __shared__ float r[128];
  r[threadIdx.x] = bk[threadIdx.x] * q[b * HH + threadIdx.x];
  __syncthreads();
  for (int s = 64; s > 0; s >>= 1) {
    if (threadIdx.x < s) r[threadIdx.x] += r[threadIdx.x + s];
    __syncthreads();
  }
  if (threadIdx.x == 0) dot0[b] = r[0];
}

// ---------------------------------------------------------------------------
// K4: dot[b,t] = x[b,t,:].qk[b] + dot0[b]; decay, e. One row per wave step.
// ---------------------------------------------------------------------------
__global__ __launch_bounds__(256) void dot_decay_kernel(
    const float* __restrict__ x, const float* __restrict__ mask,
    const float* __restrict__ qk, const float* __restrict__ dot0,
    const float* __restrict__ rate, float* __restrict__ dotOut,
    float* __restrict__ decayOut, float* __restrict__ e) {
  const int b = blockIdx.x;
  const int wave = threadIdx.x >> 5, lane = threadIdx.x & 31;
  const float4 qk4 = *(const float4*)(qk + b * DD + lane * 4);
  const float rate0 = rate[0];
  const float d0 = dot0[b];
  const int t0 = blockIdx.y * 256 + wave * 32;
  for (int i = 0; i < 32; ++i) {
    const int t = t0 + i;
    const float4* row = (const float4*)(x + (b * TT + t) * DD);
    const float4 xv = row[lane];  // 512B coalesced per wave
    float s = xv.x * qk4.x + xv.y * qk4.y + xv.z * qk4.z + xv.w * qk4.w;
    for (int off = 16; off > 0; off >>= 1) s += __shfl_xor(s, off, 32);
    if (lane == 0) {
      const float dv = s + d0;
      const float sig = 1.0f / (1.0f + expf(-dv));
      const float tm = log1pf((1.0f - sig) * mask[b * TT + t]);
      const float dec = rate0 * tm;
      dotOut[b * TT + t] = dv;
      decayOut[b * TT + t] = dec;
      e[b * TT + t] = dv - dec;
    }
  }
}

// ---------------------------------------------------------------------------
// K6: per-batch softmax over T, write a, then xa[b,d] = sum_t a[t]*x[b,t,d]
// ---------------------------------------------------------------------------
__global__ __launch_bounds__(512) void softmax_xa_kernel(const float* __restrict__ x,
                                                         const float* __restrict__ e,
                                                         float* __restrict__ aOut,
                                                         float* __restrict__ xa) {
  const int b = blockIdx.x, tid = threadIdx.x;
  __shared__ float aLds[TT];
  __shared__ float red[512];

  float ev[4];
  float m = -3.4e38f;
#pragma unroll
  for (int k = 0; k < 4; ++k) {
    ev[k] = e[b * TT + k * 512 + tid];
    m = fmaxf(m, ev[k]);
  }
  red[tid] = m;
  __syncthreads();
  for (int s = 256; s > 0; s >>= 1) {
    if (tid < s) red[tid] = fmaxf(red[tid], red[tid + s]);
    __syncthreads();
  }
  const float M = red[0];
  __syncthreads();

  float p[4];
  float ls = 0.f;
#pragma unroll
  for (int k = 0; k < 4; ++k) {
    p[k] = expf(ev[k] - M);
    ls += p[k];
  }
  red[tid] = ls;
  __syncthreads();
  for (int s = 256; s > 0; s >>= 1) {
    if (tid < s) red[tid] += red[tid + s];
    __syncthreads();
  }
  const float inv = 1.0f / red[0];
  __syncthreads();

#pragma unroll
  for (int k = 0; k < 4; ++k) {
    const float a = p[k] * inv;
    aLds[k * 512 + tid] = a;
    aOut[b * TT + k * 512 + tid] = a;
  }
  __syncthreads();

  const int ts = tid >> 7, d = tid & 127;
  float acc = 0.f;
  for (int t = ts; t < TT; t += 4) acc += aLds[t] * x[(b * TT + t) * DD + d];
  red[tid] = acc;
  __syncthreads();
  if (ts == 0) xa[b * DD + d] = red[d] + red[128 + d] + red[256 + d] + red[384 + d];
}

// ---------------------------------------------------------------------------
extern "C" void kernel_launch(void* const* d_in, const int* in_sizes, int n_in,
                              void* d_out, int out_size, void* d_ws, size_t ws_size,
                              hipStream_t stream) {
  (void)in_sizes; (void)n_in; (void)out_size; (void)ws_size;
  const float* x    = (const float*)d_in[0];
  const float* mask = (const float*)d_in[1];
  const float* Wq   = (const float*)d_in[2];
  const float* bq   = (const float*)d_in[3];
  const float* Wk   = (const float*)d_in[4];
  const float* bk   = (const float*)d_in[5];
  const float* Wv   = (const float*)d_in[6];
  const float* bv   = (const float*)d_in[7];
  const float* rate = (const float*)d_in[8];

  float* wsf     = (float*)d_ws;
  float* partial = wsf;                    // B*4*D = 65536
  float* xmean   = partial + BB * 4 * DD;  // B*D
  float* q       = xmean + BB * DD;        // B*H
  float* qk      = q + BB * HH;            // B*D
  float* dot0    = qk + BB * DD;           // B
  float* e       = dot0 + BB;              // B*T
  float* xa      = e + BB * TT;            // B*D

  float* out      = (float*)d_out;         // [B,H]
  float* aOut     = out + BB * HH;         // [B,T]
  float* dotOut   = aOut + BB * TT;        // [B,T]
  float* decayOut = dotOut + BB * TT;      // [B,T]

  // 1) xmean = mean_t(x)
  colsum_partial<<<dim3(BB, 4), 256, 0, stream>>>(x, partial);
  colsum_final<<<BB, 128, 0, stream>>>(partial, xmean);
  // 2) q = xmean @ Wq^T + bq      (WMMA fp32)
  gemm128_wmma<true><<<dim3(8, 8), 32, 0, stream>>>(xmean, Wq, bq, q);
  // 3) qk = q @ Wk                (WMMA fp32), dot0 = bk . q
  gemm128_wmma<false><<<dim3(8, 8), 32, 0, stream>>>(q, Wk, nullptr, qk);
  dot0_kernel<<<BB, 128, 0, stream>>>(q, bk, dot0);
  // 4) dot = x.qk + dot0, decay, e
  dot_decay_kernel<<<dim3(BB, 8), 256, 0, stream>>>(x, mask, qk, dot0, rate,
                                                    dotOut, decayOut, e);
  // 5) a = softmax(e); xa = sum_t a*x
  softmax_xa_kernel<<<BB, 512, 0, stream>>>(x, e, aOut, xa);
  // 6) out = xa @ Wv^T + bv       (WMMA fp32; exact because sum(a)=1)
  gemm128_wmma<true><<<dim3(8, 8), 32, 0, stream>>>(xa, Wv, bv, out);
}